// ActionPredictor_58746562674774
// MI455X (gfx1250) — compile-verified
//
#include <hip/hip_runtime.h>
#include <hip/hip_bf16.h>

typedef float v2f __attribute__((ext_vector_type(2)));
typedef float v8f __attribute__((ext_vector_type(8)));

#define WAVES_PER_BLOCK 8
#define OUT_CH 32

// ---------------- WMMA helpers (V_WMMA_F32_16X16X4_F32) ----------------
__device__ __forceinline__ v8f wmma4(v2f a, v2f b, v8f c) {
    // (neg_a, A, neg_b, B, c_mod, C, reuse_a, reuse_b)
    return __builtin_amdgcn_wmma_f32_16x16x4_f32(false, a, false, b, (short)0, c,
                                                 false, false);
}

// A fragment from a row-major [M x 32] matrix, 16x4 tile at (m0, k0).
// Lanes 0-15: {A[m][k0], A[m][k0+1]}, lanes 16-31: {A[m][k0+2], A[m][k0+3]}.
__device__ __forceinline__ v2f afrag32(const float* __restrict__ P, int m0, int lm,
                                       int half, int k0) {
    const float* p = P + (size_t)(m0 + lm) * 32 + k0 + (half << 1);
    return (v2f){p[0], p[1]};
}

// B fragment from a row-major [32 x 32] weight, 4x16 tile at (k0, col n).
// Lanes 0-15: {W[k0][n], W[k0+1][n]}, lanes 16-31: {W[k0+2][n], W[k0+3][n]}.
__device__ __forceinline__ v2f bfrag32(const float* __restrict__ W, int k0, int half,
                                       int n) {
    const float* p = W + (size_t)(k0 + (half << 1)) * 32 + n;
    return (v2f){p[0], p[32]};
}

// A fragment for K=3 (padded to 4) from row-major [M x 3].
__device__ __forceinline__ v2f afragK3(const float* __restrict__ P, int m0, int lm,
                                       int half) {
    const float* p = P + (size_t)(m0 + lm) * 3;
    return half ? (v2f){p[2], 0.0f} : (v2f){p[0], p[1]};
}

// B fragment for K=3 (padded to 4) from row-major [3 x 32].
__device__ __forceinline__ v2f bfragK3(const float* __restrict__ W, int half, int n) {
    return half ? (v2f){W[64 + n], 0.0f} : (v2f){W[n], W[32 + n]};
}

__device__ __forceinline__ float sigmoidf_(float v) {
    return 1.0f / (1.0f + __expf(-v));
}

// ---------------- Edge-phase kernels ----------------
__global__ __launch_bounds__(256) void k_degree(const int* __restrict__ src,
                                                const int* __restrict__ dst,
                                                const float* __restrict__ ew,
                                                float* __restrict__ deg, int E) {
    int e = blockIdx.x * blockDim.x + threadIdx.x;
    if (e >= E) return;
    int s = src[e], d = dst[e];
    if (s != d) atomicAdd(&deg[s], ew[e]);
}

__global__ __launch_bounds__(256) void k_dis(float* __restrict__ deg, int N) {
    int i = blockIdx.x * blockDim.x + threadIdx.x;
    if (i >= N) return;
    float dg = deg[i];
    deg[i] = (dg > 0.0f) ? rsqrtf(dg) : 0.0f;  // in-place: deg -> dis
}

__global__ __launch_bounds__(256) void k_nw_xscatter(
    const int* __restrict__ src, const int* __restrict__ dst,
    const float* __restrict__ ew, const float* __restrict__ dis,
    const float* __restrict__ x, float* __restrict__ nw,
    float* __restrict__ tx1x, int E) {
    int e = blockIdx.x * blockDim.x + threadIdx.x;
    if (e >= E) return;
    int s = src[e], d = dst[e];
    float w = (s == d) ? 0.0f : ew[e];
    float v = -dis[s] * w * dis[d];
    nw[e] = v;
    if (v != 0.0f) {
        const float* xs = x + (size_t)s * 3;
        atomicAdd(&tx1x[(size_t)d * 3 + 0], v * xs[0]);
        atomicAdd(&tx1x[(size_t)d * 3 + 1], v * xs[1]);
        atomicAdd(&tx1x[(size_t)d * 3 + 2], v * xs[2]);
    }
}

// 32 lanes per edge (lane = channel): coalesced gather of feat[src], atomic
// scatter into acc[dst]. Used for both Tx1h and Tx1hr.
__global__ __launch_bounds__(256) void k_chscatter(
    const int* __restrict__ src, const int* __restrict__ dst,
    const float* __restrict__ nw, const float* __restrict__ feat,
    float* __restrict__ acc, int E) {
    int tid = blockIdx.x * blockDim.x + threadIdx.x;
    int e = tid >> 5;
    if (e >= E) return;
    int ch = tid & 31;
    float v = nw[e];
    if (v != 0.0f) {
        int s = src[e], d = dst[e];
        atomicAdd(&acc[(size_t)d * 32 + ch], v * feat[(size_t)s * 32 + ch]);
    }
}

// ---------------- Node-phase kernel 1: gates (Z, R, hr, partial) ----------------
__global__ __launch_bounds__(256) void gconv_gates(
    const float* __restrict__ x, const float* __restrict__ h,
    const float* __restrict__ tx1x, const float* __restrict__ tx1h,
    const float* __restrict__ Wxz, const float* __restrict__ bxz,
    const float* __restrict__ Whz, const float* __restrict__ bhz,
    const float* __restrict__ Wxr, const float* __restrict__ bxr,
    const float* __restrict__ Whr, const float* __restrict__ bhr,
    const float* __restrict__ Wxh, const float* __restrict__ bxh,
    const float* __restrict__ Whh, const float* __restrict__ bhh,
    float* __restrict__ Zbuf, float* __restrict__ hrbuf,
    float* __restrict__ Pbuf, int N) {
    const int lane = threadIdx.x & 31;
    const int wid  = threadIdx.x >> 5;
    const int m0   = (blockIdx.x * WAVES_PER_BLOCK + wid) << 4;
    if (m0 >= N) return;  // wave-uniform: EXEC stays all-ones for WMMA
    const int half = lane >> 4;
    const int lm   = lane & 15;

    const v2f aX  = afragK3(x,    m0, lm, half);
    const v2f aTX = afragK3(tx1x, m0, lm, half);

    for (int nt = 0; nt < 2; ++nt) {
        const int n = (nt << 4) + lm;
        v8f accZ = {0.f, 0.f, 0.f, 0.f, 0.f, 0.f, 0.f, 0.f};
        v8f accR = accZ;
        v8f accP = accZ;

        // x-path (K=3 padded to 4): x@W0 + Tx1x@W1 for z, r, h-candidate
        accZ = wmma4(aX,  bfragK3(Wxz,      half, n), accZ);
        accZ = wmma4(aTX, bfragK3(Wxz + 96, half, n), accZ);
        accR = wmma4(aX,  bfragK3(Wxr,      half, n), accR);
        accR = wmma4(aTX, bfragK3(Wxr + 96, half, n), accR);
        accP = wmma4(aX,  bfragK3(Wxh,      half, n), accP);
        accP = wmma4(aTX, bfragK3(Wxh + 96, half, n), accP);

        // h-path (K=32): h@Whz0 + Tx1h@Whz1 ; h@Whr0 + Tx1h@Whr1
#pragma unroll
        for (int k0 = 0; k0 < 32; k0 += 4) {
            v2f ah = afrag32(h,    m0, lm, half, k0);
            v2f at = afrag32(tx1h, m0, lm, half, k0);
            accZ = wmma4(ah, bfrag32(Whz,        k0, half, n), accZ);
            accZ = wmma4(at, bfrag32(Whz + 1024, k0, half, n), accZ);
            accR = wmma4(ah, bfrag32(Whr,        k0, half, n), accR);
            accR = wmma4(at, bfrag32(Whr + 1024, k0, half, n), accR);
        }

        const float bZ = bxz[n] + bhz[n];
        const float bR = bxr[n] + bhr[n];
        const float bP = bxh[n] + bhh[n];
#pragma unroll
        for (int i = 0; i < 8; ++i) {
            int row = m0 + i + (half << 3);
            size_t idx = (size_t)row * 32 + n;
            float z  = sigmoidf_(accZ[i] + bZ);
            float r  = sigmoidf_(accR[i] + bR);
            float hv = h[idx];
            Zbuf[idx]  = z;
            hrbuf[idx] = hv * r;
            Pbuf[idx]  = accP[i] + bP;
        }
    }
}

// ---------------- Node-phase kernel 2: H_tilde, h_new, linear out ----------------
__global__ __launch_bounds__(256) void gconv_final(
    const float* __restrict__ h, const float* __restrict__ hrb,
    const float* __restrict__ tx1hr, const float* __restrict__ Whh,
    const float* __restrict__ Zbuf, const float* __restrict__ Pbuf,
    const float* __restrict__ Wlin, const float* __restrict__ blin,
    float* __restrict__ out, float* __restrict__ hnew, int N) {
    __shared__ float lds[WAVES_PER_BLOCK][16 * 33];
    const int lane = threadIdx.x & 31;
    const int wid  = threadIdx.x >> 5;
    const int m0   = (blockIdx.x * WAVES_PER_BLOCK + wid) << 4;
    if (m0 >= N) return;
    const int half = lane >> 4;
    const int lm   = lane & 15;

    for (int nt = 0; nt < 2; ++nt) {
        const int n = (nt << 4) + lm;
        v8f acc = {0.f, 0.f, 0.f, 0.f, 0.f, 0.f, 0.f, 0.f};
#pragma unroll
        for (int k0 = 0; k0 < 32; k0 += 4) {
            acc = wmma4(afrag32(hrb,   m0, lm, half, k0),
                        bfrag32(Whh,        k0, half, n), acc);
            acc = wmma4(afrag32(tx1hr, m0, lm, half, k0),
                        bfrag32(Whh + 1024, k0, half, n), acc);
        }
#pragma unroll
        for (int i = 0; i < 8; ++i) {
            int row = m0 + i + (half << 3);
            size_t idx = (size_t)row * 32 + n;
            float ht = tanhf(acc[i] + Pbuf[idx]);
            float z  = Zbuf[idx];
            float hn = z * h[idx] + (1.0f - z) * ht;
            hnew[idx] = hn;
            lds[wid][(i + (half << 3)) * 33 + n] = hn;
        }
    }

    // Same-wave LDS write->read ordering (CDNA5 split counters).
    asm volatile("s_wait_dscnt 0" ::: "memory");

    if (lane < 16) {
        const float* row = &lds[wid][lane * 33];
        float s = 0.0f;
#pragma unroll
        for (int c = 0; c < OUT_CH; ++c) s += row[c] * Wlin[c];
        out[m0 + lane] = s + blin[0];
    }
}

// ---------------- Launcher ----------------
extern "C" void kernel_launch(void* const* d_in, const int* in_sizes, int n_in,
                              void* d_out, int out_size, void* d_ws, size_t ws_size,
                              hipStream_t stream) {
    const float* x    = (const float*)d_in[0];
    const int*   ei   = (const int*)d_in[1];
    const float* ew   = (const float*)d_in[2];
    const float* h    = (const float*)d_in[3];
    const float* Wxz  = (const float*)d_in[4];
    const float* bxz  = (const float*)d_in[5];
    const float* Whz  = (const float*)d_in[6];
    const float* bhz  = (const float*)d_in[7];
    const float* Wxr  = (const float*)d_in[8];
    const float* bxr  = (const float*)d_in[9];
    const float* Whr  = (const float*)d_in[10];
    const float* bhr  = (const float*)d_in[11];
    const float* Wxh  = (const float*)d_in[12];
    const float* bxh  = (const float*)d_in[13];
    const float* Whh  = (const float*)d_in[14];
    const float* bhh  = (const float*)d_in[15];
    const float* Wlin = (const float*)d_in[16];
    const float* blin = (const float*)d_in[17];

    const int N = in_sizes[0] / 3;       // 100000 (multiple of 16)
    const int E = in_sizes[2];           // 1600000
    const int* src = ei;
    const int* dst = ei + E;

    // Workspace layout (floats): deg/dis[N] | nw[E] | Tx1x[3N] | Tx1h[32N]
    //                            | hr[32N] | Z[32N] | partial[32N]
    float* ws    = (float*)d_ws;
    float* deg   = ws;
    float* nw    = deg + N;
    float* tx1x  = nw + E;
    float* tx1h  = tx1x + (size_t)3 * N;   // reused as Tx1hr after gates
    float* hrb   = tx1h + (size_t)32 * N;
    float* Zb    = hrb + (size_t)32 * N;
    float* Pb    = Zb + (size_t)32 * N;

    float* out  = (float*)d_out;     // [N]
    float* hnew = out + N;           // [N,32]

    const int tpb = 256;
    const int eBlocks = (E + tpb - 1) / tpb;
    const int nBlocks = (N + tpb - 1) / tpb;
    const long total32 = (long)E * 32;
    const int sBlocks = (int)((total32 + tpb - 1) / tpb);
    const int gBlocks = (N + 16 * WAVES_PER_BLOCK - 1) / (16 * WAVES_PER_BLOCK);

    hipMemsetAsync(deg, 0, sizeof(float) * (size_t)N, stream);
    hipMemsetAsync(tx1x, 0, sizeof(float) * (size_t)3 * N, stream);
    hipMemsetAsync(tx1h, 0, sizeof(float) * (size_t)32 * N, stream);

    k_degree<<<eBlocks, tpb, 0, stream>>>(src, dst, ew, deg, E);
    k_dis<<<nBlocks, tpb, 0, stream>>>(deg, N);
    k_nw_xscatter<<<eBlocks, tpb, 0, stream>>>(src, dst, ew, deg, x, nw, tx1x, E);
    k_chscatter<<<sBlocks, tpb, 0, stream>>>(src, dst, nw, h, tx1h, E);

    gconv_gates<<<gBlocks, tpb, 0, stream>>>(x, h, tx1x, tx1h,
                                             Wxz, bxz, Whz, bhz,
                                             Wxr, bxr, Whr, bhr,
                                             Wxh, bxh, Whh, bhh,
                                             Zb, hrb, Pb, N);

    hipMemsetAsync(tx1h, 0, sizeof(float) * (size_t)32 * N, stream);  // -> Tx1hr
    k_chscatter<<<sBlocks, tpb, 0, stream>>>(src, dst, nw, hrb, tx1h, E);

    gconv_final<<<gBlocks, tpb, 0, stream>>>(h, hrb, tx1h, Whh, Zb, Pb,
                                             Wlin, blin, out, hnew, N);
}